// SparseConvTransitionBlock_38981123179036
// MI455X (gfx1250) — compile-verified
//
#include <hip/hip_runtime.h>

#define N_IN   200000
#define N_OUTP 100000
#define EDGES  1600000
#define CIN    64
#define KK     9
#define C_A    96
#define C_B    32
#define KDIM   (KK * CIN)      // 576 = GEMM K dimension
#define LSTR   580             // LDS row stride (580 % 64 == 4 -> conflict-free b64 A-frag loads)
#define EPSF   1e-8f

typedef __attribute__((ext_vector_type(2))) float v2f;
typedef __attribute__((ext_vector_type(8))) float v8f;

// --- Kernel 1: CSR row starts via binary search on the sorted out-index ---
__global__ __launch_bounds__(256) void row_starts_kernel(
    const int* __restrict__ oidx, int* __restrict__ rs) {
  int n = blockIdx.x * 256 + threadIdx.x;
  if (n > N_OUTP) return;
  int lo = 0, hi = EDGES;
  while (lo < hi) {
    int mid = (lo + hi) >> 1;
    if (oidx[mid] < n) lo = mid + 1; else hi = mid;
  }
  rs[n] = lo;
}

// --- Kernel 1b: transpose W to [Cout, KDIM] so B fragments are contiguous b64 loads ---
__global__ __launch_bounds__(256) void transpose_w_kernel(
    const float* __restrict__ Wa, const float* __restrict__ Wb,
    float* __restrict__ WtA, float* __restrict__ WtB) {
  int i = blockIdx.x * 256 + threadIdx.x;
  if (i < KDIM * C_A) {
    int k = i / C_A, o = i % C_A;
    WtA[o * KDIM + k] = Wa[i];
  } else if (i < KDIM * (C_A + C_B)) {
    int j = i - KDIM * C_A;
    int k = j / C_B, o = j % C_B;
    WtB[o * KDIM + k] = Wb[j];
  }
}

// --- Kernel 2: bin 16 output points into LDS, WMMA-contract with W, fuse epilogue ---
__global__ __launch_bounds__(256) void spconv_fused_kernel(
    const float* __restrict__ feats, const float* __restrict__ importance,
    const float* __restrict__ WtA, const float* __restrict__ ba,
    const float* __restrict__ WtB, const float* __restrict__ bb,
    const int* __restrict__ nbr, const int* __restrict__ kidx,
    const int* __restrict__ oidx, const int* __restrict__ rs,
    float* __restrict__ out) {
  // bins: [0 .. 16*LSTR)  = plain sums (conv1a)
  //       [16*LSTR .. 2*16*LSTR) = importance-weighted sums (conv1b)
  __shared__ float sbins[2 * 16 * LSTR];
  __shared__ float simp[16];
  __shared__ float sfac[32];   // [0..15]=1.0 (conv1a rows), [16..31]=1/max(imp,eps)

  const int tid  = threadIdx.x;
  const int base = blockIdx.x * 16;           // first output point of this tile

  for (int i = tid; i < 2 * 16 * LSTR; i += 256) sbins[i] = 0.0f;
  if (tid < 16) simp[tid] = 0.0f;
  __syncthreads();

  // --- Edge accumulation: 16 lanes per edge, 4 channels (float4) per lane ---
  const int estart = rs[base];
  const int eend   = rs[base + 16];
  const int g   = tid >> 4;                   // edge group 0..15
  const int lig = tid & 15;                   // lane in group -> channels 4*lig..4*lig+3
  for (int e = estart + g; e < eend; e += 16) {
    const int   src = nbr[e];
    const int   kk  = kidx[e];
    const int   row = oidx[e] - base;         // 0..15 by sortedness
    const float imp = importance[src];
    const float4 v  = ((const float4*)(feats + (size_t)src * CIN))[lig];
    float* pa = &sbins[row * LSTR + kk * CIN + lig * 4];
    float* pb = pa + 16 * LSTR;
    atomicAdd(pa + 0, v.x);       atomicAdd(pa + 1, v.y);
    atomicAdd(pa + 2, v.z);       atomicAdd(pa + 3, v.w);
    atomicAdd(pb + 0, imp * v.x); atomicAdd(pb + 1, imp * v.y);
    atomicAdd(pb + 2, imp * v.z); atomicAdd(pb + 3, imp * v.w);
    if (lig == 0) atomicAdd(&simp[row], imp);
  }
  __syncthreads();

  // second output: raw imp_sum; build normalize-factor table once per block
  if (tid < 16) {
    const float s = simp[tid];
    out[(size_t)N_OUTP * 128 + base + tid] = s;
    sfac[tid]      = 1.0f;                    // conv1a: no normalization
    sfac[16 + tid] = 1.0f / fmaxf(s, EPSF);   // conv1b: importance normalize
  }
  __syncthreads();

  // --- WMMA contraction: D[16 x 16] += A[16 x 4] * B[4 x 16], K = 576 ---
  const int w    = tid >> 5;                  // wave 0..7 (wave32)
  const int lane = tid & 31;
  const int half = lane >> 4;                 // 0: K+0..1 | 1: K+2..3 (f32 A/B-frag layout)
  const int m    = lane & 15;                 // M (A) / N (B) index

  const bool isB = (w >= 6);                  // waves 6,7 -> conv1b tiles
  const float* Wt   = isB ? WtB : WtA;        // pre-transposed [Cout, KDIM]
  const float* bias = isB ? bb : ba;
  const int nb   = isB ? (w - 6) * 16 : w * 16;         // column tile within its conv
  const int colg = isB ? (C_A + (w - 6) * 16) : w * 16; // column in concat output
  const int lcol = nb + m;                    // local output channel
  const float* abin = sbins + (isB ? 16 * LSTR : 0) + m * LSTR + 2 * half;
  const float* brow = Wt + (size_t)lcol * KDIM + 2 * half;

  v8f c = {};
  #pragma unroll 4
  for (int kg = 0; kg < KDIM; kg += 4) {
    v2f a = *(const v2f*)(abin + kg);         // ds_load_b64, conflict-free
    v2f b = *(const v2f*)(brow + kg);         // single global_load_b64 (contiguous)
    c = __builtin_amdgcn_wmma_f32_16x16x4_f32(
        /*neg_a=*/false, a, /*neg_b=*/false, b,
        /*c_mod=*/(short)0, c, /*reuse_a=*/false, /*reuse_b=*/false);
  }

  // --- Epilogue: unconditional normalize + bias + ReLU, store concat [N_out,128] ---
  const float bv = bias[lcol];
  const float* facp = sfac + (isB ? 16 : 0) + 8 * half; // factors for this lane's 8 rows
  const float4 f0 = *(const float4*)(facp);
  const float4 f1 = *(const float4*)(facp + 4);
  const float fr[8] = {f0.x, f0.y, f0.z, f0.w, f1.x, f1.y, f1.z, f1.w};
  #pragma unroll
  for (int r = 0; r < 8; ++r) {
    const int rowl = r + 8 * half;            // C/D layout: VGPR r -> M = r (+8 for hi lanes)
    const float val = fmaxf(c[r] * fr[r] + bv, 0.0f);
    out[(size_t)(base + rowl) * 128 + colg + m] = val;
  }
}

extern "C" void kernel_launch(void* const* d_in, const int* in_sizes, int n_in,
                              void* d_out, int out_size, void* d_ws, size_t ws_size,
                              hipStream_t stream) {
  const float* feats      = (const float*)d_in[0];
  const float* importance = (const float*)d_in[1];
  const float* Wa         = (const float*)d_in[2];
  const float* ba         = (const float*)d_in[3];
  const float* Wb         = (const float*)d_in[4];
  const float* bb         = (const float*)d_in[5];
  const int*   nbr        = (const int*)d_in[6];
  const int*   kidx       = (const int*)d_in[7];
  const int*   oidx       = (const int*)d_in[8];
  float* out = (float*)d_out;

  // Workspace layout: rs[(N_OUTP+1) ints] | WtA [96*576 f32] | WtB [32*576 f32]
  int*   rs  = (int*)d_ws;
  size_t off = (((size_t)(N_OUTP + 1) * sizeof(int)) + 255) & ~(size_t)255;
  float* WtA = (float*)((char*)d_ws + off);
  float* WtB = WtA + (size_t)C_A * KDIM;

  row_starts_kernel<<<(N_OUTP + 1 + 255) / 256, 256, 0, stream>>>(oidx, rs);
  transpose_w_kernel<<<(KDIM * (C_A + C_B) + 255) / 256, 256, 0, stream>>>(Wa, Wb, WtA, WtB);
  spconv_fused_kernel<<<N_OUTP / 16, 256, 0, stream>>>(
      feats, importance, WtA, ba, WtB, bb, nbr, kidx, oidx, rs, out);
}